// GraphProjection_53936199303568
// MI455X (gfx1250) — compile-verified
//
#include <hip/hip_runtime.h>
#include <hip/hip_bf16.h>

// CDNA5 / gfx1250 wave32 WMMA types
typedef __attribute__((ext_vector_type(2))) float v2f;
typedef __attribute__((ext_vector_type(4))) float v4f;
typedef __attribute__((ext_vector_type(8))) float v8f;

#define OUT_STRIDE 451
#define NWAVES 8                 // waves per block (256 threads)
#define GSIZE  4                 // waves cooperating on one 32-query group
#define NGROUP (NWAVES / GSIZE)  // query groups per block = 2
#define TPITCH 20                // floats; 80B rows -> 16B-aligned b128 reads

#if defined(__has_builtin)
#if __has_builtin(__builtin_amdgcn_wmma_f32_16x16x4_f32)
#define HAVE_WMMA_F32_K4 1
#endif
#endif

// Block = 8 waves. Waves [g*4 .. g*4+3] co-own query group g (32 queries):
// each wave scans M/4 points with V_WMMA_F32_16X16X4_F32 score tiles
// (score = |p|^2 - 2 q.p, same ordering as squared distance), keeps a
// register-resident sorted top-8 behind a branch-free min16 early-out, then
// the 4 partial lists are merged and the feature gather columns split 4 ways.
__global__ __launch_bounds__(NWAVES * 32) void knn_stage_kernel(
    const float* __restrict__ verts,   // [N,3]
    const float* __restrict__ pc,      // [3,M]
    const float* __restrict__ feat,    // [Df,M]
    float* __restrict__ out,           // [N,OUT_STRIDE]
    int M, int Df, int colOff)
{
  __shared__ __align__(16) float sTile[NWAVES][32][TPITCH];
  __shared__ float sPS[NGROUP][GSIZE][32][8];
  __shared__ int   sPI[NGROUP][GSIZE][32][8];
#if !defined(HAVE_WMMA_F32_K4)
  __shared__ float sVert[NGROUP][32][4];
#endif

  const int lane  = threadIdx.x & 31;
  const int wave  = threadIdx.x >> 5;
  const int grp   = wave / GSIZE;   // query group within block
  const int slot  = wave % GSIZE;   // which M chunk this wave scans
  const int qbase = blockIdx.x * (NGROUP * 32) + grp * 32;
  const int q     = qbase + lane;   // this lane's query

  const float qx = verts[q * 3 + 0];
  const float qy = verts[q * 3 + 1];
  const float qz = verts[q * 3 + 2];

  const int nIdx  = lane & 15;      // column within 16-wide tile
  const int khalf = lane >> 4;      // lanes 0-15 -> K={0,1}, 16-31 -> K={2,3}

#if defined(HAVE_WMMA_F32_K4)
  // A fragments (16x4 f32 layout). Row = (-2x, -2y, -2z, 1).
  const int m0 = qbase + nIdx;
  const int m1 = qbase + 16 + nIdx;
  v2f a0, a1;
  {
    const float x0 = verts[m0 * 3 + 0], y0 = verts[m0 * 3 + 1], z0 = verts[m0 * 3 + 2];
    const float x1 = verts[m1 * 3 + 0], y1 = verts[m1 * 3 + 1], z1 = verts[m1 * 3 + 2];
    a0.x = khalf ? (-2.0f * z0) : (-2.0f * x0);
    a0.y = khalf ? 1.0f         : (-2.0f * y0);
    a1.x = khalf ? (-2.0f * z1) : (-2.0f * x1);
    a1.y = khalf ? 1.0f         : (-2.0f * y1);
  }
#else
  sVert[grp][lane][0] = qx;
  sVert[grp][lane][1] = qy;
  sVert[grp][lane][2] = qz;
  __syncthreads();
#endif

  // register-resident top-8 (sorted ascending by score)
  float bd[8];
  int   bi[8];
#pragma unroll
  for (int k = 0; k < 8; ++k) { bd[k] = 3.0e38f; bi[k] = 0; }

  const int chunk = M / GSIZE;          // divisible: M in {32768,16384,8192}
  const int cbase = slot * chunk;
  const int cend  = cbase + chunk;

  for (int base = cbase; base < cend; base += 16) {
    const int   n  = base + nIdx;
    const float px = pc[n];
    const float py = pc[M + n];
    const float pz = pc[2 * M + n];
    const float p2 = px * px + py * py + pz * pz;

#if defined(HAVE_WMMA_F32_K4)
    // B fragment (4x16 f32). Col = (px, py, pz, |p|^2). Shared by both tiles.
    v2f b;
    b.x = khalf ? pz : px;
    b.y = khalf ? p2 : py;
    v8f cz = {};
    v8f c0 = __builtin_amdgcn_wmma_f32_16x16x4_f32(false, a0, false, b,
                                                   (short)0, cz, false, false);
    v8f c1 = __builtin_amdgcn_wmma_f32_16x16x4_f32(false, a1, false, b,
                                                   (short)0, cz, false, false);
    // C layout: VGPR r, lane L -> row (r + 8*(L>>4)), col (L&15)
#pragma unroll
    for (int r = 0; r < 8; ++r) {
      sTile[wave][r + 8 * khalf][nIdx]      = c0[r];
      sTile[wave][16 + r + 8 * khalf][nIdx] = c1[r];
    }
#else
#pragma unroll
    for (int r = 0; r < 8; ++r) {
      const int row = r + 8 * khalf;
      const float vx0 = sVert[grp][row][0],      vy0 = sVert[grp][row][1],      vz0 = sVert[grp][row][2];
      const float vx1 = sVert[grp][16 + row][0], vy1 = sVert[grp][16 + row][1], vz1 = sVert[grp][16 + row][2];
      sTile[wave][row][nIdx]      = p2 - 2.0f * (vx0 * px + vy0 * py + vz0 * pz);
      sTile[wave][16 + row][nIdx] = p2 - 2.0f * (vx1 * px + vy1 * py + vz1 * pz);
    }
#endif
    // Intra-wave LDS transpose only: CDNA5 keeps same-wave DS ops in order, so
    // no hardware barrier is needed -- just stop the compiler from reordering.
    __builtin_amdgcn_wave_barrier();

    // lane owns query row `lane`: read 16 scores as 4x b128
    const v4f* rowp = (const v4f*)(&sTile[wave][lane][0]);
    const v4f r0 = rowp[0];
    const v4f r1 = rowp[1];
    const v4f r2 = rowp[2];
    const v4f r3 = rowp[3];
    __builtin_amdgcn_wave_barrier();   // WAR vs next iteration's stores

    // branch-free min16 early-out: most chunks contain no top-8 candidate
    const float ma = fminf(fminf(r0.x, r0.y), fminf(r0.z, r0.w));
    const float mb = fminf(fminf(r1.x, r1.y), fminf(r1.z, r1.w));
    const float mc = fminf(fminf(r2.x, r2.y), fminf(r2.z, r2.w));
    const float md = fminf(fminf(r3.x, r3.y), fminf(r3.z, r3.w));
    const float m16 = fminf(fminf(ma, mb), fminf(mc, md));

    if (m16 < bd[7]) {
      const float cand[16] = {r0.x, r0.y, r0.z, r0.w, r1.x, r1.y, r1.z, r1.w,
                              r2.x, r2.y, r2.z, r2.w, r3.x, r3.y, r3.z, r3.w};
#pragma unroll
      for (int j = 0; j < 16; ++j) {
        const float s = cand[j];
        if (s < bd[7]) {
          bd[7] = s;
          bi[7] = base + j;
#pragma unroll
          for (int t = 7; t > 0; --t) {
            if (bd[t] < bd[t - 1]) {
              const float td = bd[t]; bd[t] = bd[t - 1]; bd[t - 1] = td;
              const int   ti = bi[t]; bi[t] = bi[t - 1]; bi[t - 1] = ti;
            }
          }
        }
      }
    }
  }

  // publish this wave's partial top-8
#pragma unroll
  for (int k = 0; k < 8; ++k) {
    sPS[grp][slot][lane][k] = bd[k];
    sPI[grp][slot][lane][k] = bi[k];
  }
  __syncthreads();

  // merge the group's 4 partial lists (all 4 waves compute identical result;
  // slot-major order keeps lowest-index-wins tie behavior)
#pragma unroll
  for (int k = 0; k < 8; ++k) { bd[k] = 3.0e38f; bi[k] = 0; }
#pragma unroll
  for (int sIt = 0; sIt < GSIZE; ++sIt) {
#pragma unroll
    for (int k = 0; k < 8; ++k) {
      const float s = sPS[grp][sIt][lane][k];
      const int   i = sPI[grp][sIt][lane][k];
      if (s < bd[7]) {
        bd[7] = s;
        bi[7] = i;
#pragma unroll
        for (int t = 7; t > 0; --t) {
          if (bd[t] < bd[t - 1]) {
            const float td = bd[t]; bd[t] = bd[t - 1]; bd[t - 1] = td;
            const int   ti = bi[t]; bi[t] = bi[t - 1]; bi[t - 1] = ti;
          }
        }
      }
    }
  }

  // exact squared distances for the 8 winners -> inverse-square-dist weights
  float w[8];
#pragma unroll
  for (int k = 0; k < 8; ++k) {
    const int   i  = bi[k];
    const float dx = pc[i] - qx;
    const float dy = pc[M + i] - qy;
    const float dz = pc[2 * M + i] - qz;
    w[k] = 1.0f / (1.0f + dx * dx + dy * dy + dz * dz);
  }

  // weighted mean of gathered feature rows; the 4 waves of the group split Df
  float* orow = out + (size_t)q * OUT_STRIDE + colOff;
  for (int d = slot; d < Df; d += GSIZE) {
    const float* frow = feat + (size_t)d * M;
    float acc = 0.0f;
#pragma unroll
    for (int k = 0; k < 8; ++k) acc += w[k] * frow[bi[k]];
    orow[d] = acc * 0.125f;
  }
}

__global__ void copy_inputs_kernel(const float* __restrict__ in,
                                   float* __restrict__ out, int n3)
{
  const int t = blockIdx.x * blockDim.x + threadIdx.x;
  if (t < n3) {
    const int r = t / 3;
    const int c = t - 3 * r;
    out[(size_t)r * OUT_STRIDE + c] = in[t];
  }
}

extern "C" void kernel_launch(void* const* d_in, const int* in_sizes, int n_in,
                              void* d_out, int out_size, void* d_ws, size_t ws_size,
                              hipStream_t stream) {
  const float* verts = (const float*)d_in[0];
  // d_in[1] = pc0_coords: stage-0 output is discarded by the reference -> skip.
  const float* pc1c = (const float*)d_in[2];
  const float* pc1f = (const float*)d_in[3];
  const float* pc2c = (const float*)d_in[4];
  const float* pc2f = (const float*)d_in[5];
  const float* pc3c = (const float*)d_in[6];
  const float* pc3f = (const float*)d_in[7];
  float* out = (float*)d_out;

  const int N  = in_sizes[0] / 3;     // 4096
  const int M1 = in_sizes[2] / 3;     // 32768
  const int D1 = in_sizes[3] / M1;    // 64
  const int M2 = in_sizes[4] / 3;     // 16384
  const int D2 = in_sizes[5] / M2;    // 128
  const int M3 = in_sizes[6] / 3;     // 8192
  const int D3 = in_sizes[7] / M3;    // 256
  (void)D3; (void)n_in; (void)out_size; (void)d_ws; (void)ws_size;

  {
    const int n3 = N * 3;
    copy_inputs_kernel<<<(n3 + 255) / 256, 256, 0, stream>>>(verts, out, n3);
  }

  dim3 block(NWAVES * 32);
  dim3 grid(N / (NGROUP * 32));       // 64 blocks per stage
  knn_stage_kernel<<<grid, block, 0, stream>>>(verts, pc1c, pc1f, out, M1, D1, 3);
  knn_stage_kernel<<<grid, block, 0, stream>>>(verts, pc2c, pc2f, out, M2, D2, 3 + D1);
  knn_stage_kernel<<<grid, block, 0, stream>>>(verts, pc3c, pc3f, out, M3, D3, 3 + D1 + D2);
}